// GINDecor_24962349924640
// MI455X (gfx1250) — compile-verified
//
#include <hip/hip_runtime.h>
#include <hip/hip_bf16.h>

// ---------------------------------------------------------------------------
// GINDecor on MI455X (gfx1250): gated-adjacency GCN.
//  - f32 WMMA 16x16x4 for the two A@X propagations (bit-faithful to f32 ref)
//  - TDM (tensor_load_to_lds, 6-arg builtin) double-buffered A-tile staging
//  - B staged transposed in LDS so every fragment is one ds_load_b64
// ---------------------------------------------------------------------------

typedef __attribute__((ext_vector_type(2))) float v2f;
typedef __attribute__((ext_vector_type(8))) float v8f;
typedef __attribute__((ext_vector_type(4))) unsigned int u32x4;
typedef __attribute__((ext_vector_type(8))) int i32x8;
typedef __attribute__((ext_vector_type(4))) int i32x4;

constexpr int kN   = 4096;
constexpr int kIN  = 256;
constexpr int kHID = 128;

// GEMM tiling
constexpr int BM = 64;
constexpr int BN = 64;
constexpr int KC = 32;
constexpr int LDA_S = KC + 4;  // 36 floats: 16B-aligned, conflict-free fragment reads
constexpr int LDKT  = KC + 4;  // 36 floats: B transposed [BN][KC]

#if defined(__gfx1250__) && __has_builtin(__builtin_amdgcn_tensor_load_to_lds) && \
    __has_builtin(__builtin_amdgcn_s_wait_tensorcnt)
#define USE_TDM 1
#else
#define USE_TDM 0
#endif

#if USE_TDM
// Issue one TDM 2D tile load: BM x KC f32 tile out of the kN x kN matrix,
// landing in LDS with a 36-float row stride (pad 4 DWORDs every 32 DWORDs).
__device__ __forceinline__ void tdm_load_A_tile(const float* gsrc, unsigned lds_byte_off) {
    unsigned long long ga = (unsigned long long)(uintptr_t)gsrc;
    u32x4 g0;
    g0.x = 1u;                                    // count=1 (valid user descriptor)
    g0.y = lds_byte_off;                          // LDS dest byte address
    g0.z = (unsigned)ga;                          // global_addr[31:0]
    g0.w = (unsigned)((ga >> 32) & 0x1FFFFFFull)  // global_addr[56:32]
           | (2u << 30);                          // type = 2 ("image")
    i32x8 g1;
    g1[0] = (2 << 16)      // data_size = 4 bytes
          | (1 << 20)      // pad_enable
          | (4 << 22)      // pad_interval: 32 DWORDs
          | (3 << 25);     // pad_amount: 4 DWORDs  -> LDS row stride = 36 floats
    g1[1] = (int)(((unsigned)kN & 0xFFFFu) << 16);                    // tensor_dim0 lo
    g1[2] = (int)((((unsigned)kN >> 16) & 0xFFFFu) |
                  (((unsigned)kN & 0xFFFFu) << 16));                  // dim0 hi | dim1 lo
    g1[3] = (int)((((unsigned)kN >> 16) & 0xFFFFu) | ((unsigned)KC << 16)); // dim1 hi | tile_dim0
    g1[4] = BM;                                   // tile_dim1 = BM, tile_dim2 = 0
    g1[5] = kN;                                   // tensor_dim0_stride = 4096 elements
    g1[6] = 0;
    g1[7] = 0;
    i32x4 gz4 = {0, 0, 0, 0};                     // groups 2/3 unused (2D tensor)
    i32x8 gz8 = {0, 0, 0, 0, 0, 0, 0, 0};
    // 6-arg toolchain variant: (g0, g1, g2, g3, extra, cpol)
    __builtin_amdgcn_tensor_load_to_lds(g0, g1, gz4, gz4, gz8, 0);
}
#endif

// --------------------------- K1: degrees + zero cs --------------------------
__global__ __launch_bounds__(256)
void deg_kernel(const float* __restrict__ xdeg, const float* __restrict__ ydeg,
                float* __restrict__ degr, float* __restrict__ degc,
                float* __restrict__ cs) {
    int i = blockIdx.x * 256 + threadIdx.x;
    if (i < kN) {
        degr[i] = xdeg[(size_t)i * kN];  // xdeg[i,j] == deg_row[i]
        degc[i] = ydeg[i];               // ydeg[i,j] == deg_col[j]
        cs[i] = 0.f;
    }
}

// ------------------- K2: gated adjacency A = adj*gate + I, row sums --------
__global__ __launch_bounds__(256)
void build_A_kernel(const float* __restrict__ adj,
                    const float* __restrict__ degr, const float* __restrict__ degc,
                    const float* __restrict__ We1, const float* __restrict__ be1,
                    const float* __restrict__ We2, const float* __restrict__ be2,
                    float* __restrict__ A, float* __restrict__ rs) {
    __shared__ float hbase[16], wdc[16], w2a[16], w2b[16], be2s[2];
    __shared__ float red[256];
    const int i = blockIdx.x;
    const int tid = threadIdx.x;
    const float dr = degr[i];
    if (tid < 16) {
        hbase[tid] = We1[tid] + We1[16 + tid] * dr + be1[tid];
        wdc[tid]   = We1[32 + tid];
        w2a[tid]   = We2[2 * tid + 0];
        w2b[tid]   = We2[2 * tid + 1];
    }
    if (tid < 2) be2s[tid] = be2[tid];
    __syncthreads();

    float sum = 0.f;
    for (int j = tid; j < kN; j += 256) {
        float a = adj[(size_t)i * kN + j];
        float val = 0.f;
        if (a != 0.f) {
            const float dc = degc[j];
            float l0 = be2s[0], l1 = be2s[1];
#pragma unroll
            for (int w = 0; w < 16; ++w) {
                float h = fmaxf(hbase[w] + wdc[w] * dc, 0.f);
                l0 = fmaf(h, w2a[w], l0);
                l1 = fmaf(h, w2b[w], l1);
            }
            val = a * (1.f / (1.f + __expf(l0 - l1)));  // softmax[1]
        }
        if (j == i) val += 1.f;  // self loop
        A[(size_t)i * kN + j] = val;
        sum += val;
    }
    red[tid] = sum;
    __syncthreads();
    for (int s = 128; s > 0; s >>= 1) {
        if (tid < s) red[tid] += red[tid + s];
        __syncthreads();
    }
    if (tid == 0) rs[i] = red[0];
}

// --------------------------- K3: column sums -------------------------------
__global__ __launch_bounds__(256)
void colsum_kernel(const float* __restrict__ A, float* __restrict__ cs) {
    const int j = blockIdx.x * 256 + threadIdx.x;
    const int i0 = blockIdx.y * (kN / 32);
    float sum = 0.f;
    for (int i = i0; i < i0 + kN / 32; ++i) sum += A[(size_t)i * kN + j];
    atomicAdd(&cs[j], sum);
}

// --------------------------- K4: D^{-1/2} ----------------------------------
__global__ __launch_bounds__(256)
void norm_kernel(const float* __restrict__ rs, const float* __restrict__ cs,
                 float* __restrict__ drow, float* __restrict__ dcol) {
    int i = blockIdx.x * 256 + threadIdx.x;
    if (i < kN) {
        float r = rs[i], c = cs[i];
        drow[i] = r > 0.f ? rsqrtf(r) : 0.f;
        dcol[i] = c > 0.f ? rsqrtf(c) : 0.f;
    }
}

// --------------------------- K4b: xs = dcol (.) x --------------------------
__global__ __launch_bounds__(256)
void scalex_kernel(const float* __restrict__ x, const float* __restrict__ dcol,
                   float* __restrict__ xs) {
    int idx = blockIdx.x * 256 + threadIdx.x;
    int j = idx >> 8;  // / kIN (==256)
    xs[idx] = dcol[j] * x[idx];
}

// ---------- K5/K7: C = drow (.) (A @ B) + resid * X0 via f32 WMMA ----------
template <int NCOLS>
__global__ __launch_bounds__(256)
void gemm_prop_kernel(const float* __restrict__ A, const float* __restrict__ B,
                      const float* __restrict__ X0, const float* __restrict__ drow,
                      const float* __restrict__ residp, float* __restrict__ C) {
#if USE_TDM
    __shared__ float sA[2][BM * LDA_S];   // double-buffered TDM destination
#else
    __shared__ float sA[1][BM * LDA_S];
#endif
    __shared__ float sBt[BN * LDKT];      // B transposed: [n][k]

    const int tid = threadIdx.x;
    const int lane = tid & 31;
    const int wid = tid >> 5;             // 8 waves
    const int rowBase = blockIdx.y * BM;
    const int nBase = blockIdx.x * BN;
    const int rw = (wid & 3) * 16;        // wave's 16-row strip
    const int ch = (wid >> 2) * 32;       // wave's 32-col half
    const int lm = lane & 15;
    const int lhi = lane >> 4;            // 0 -> K{0,1}, 1 -> K{2,3}

    v8f acc[2] = {};

#if USE_TDM
    if (wid == 0)  // one wave drives the DMA engine
        tdm_load_A_tile(&A[(size_t)rowBase * kN], (unsigned)(uintptr_t)&sA[0][0]);
#endif

    for (int k0 = 0, buf = 0; k0 < kN; k0 += KC, buf ^= 1) {
        // Stage B tile (KC x BN) TRANSPOSED into sBt[n*LDKT + k].
#pragma unroll
        for (int it = 0; it < (KC * BN) / (4 * 256); ++it) {
            int u = tid + it * 256;
            int kr = u >> 4;
            int c4 = (u & 15) << 2;
            float4 v = *(const float4*)&B[(size_t)(k0 + kr) * NCOLS + nBase + c4];
            sBt[(c4 + 0) * LDKT + kr] = v.x;
            sBt[(c4 + 1) * LDKT + kr] = v.y;
            sBt[(c4 + 2) * LDKT + kr] = v.z;
            sBt[(c4 + 3) * LDKT + kr] = v.w;
        }
#if USE_TDM
        if (wid == 0) {
            if (k0 + KC < kN) {
                // Kick off next A tile, then wait for the current one (<=1 in flight).
                tdm_load_A_tile(&A[(size_t)rowBase * kN + k0 + KC],
                                (unsigned)(uintptr_t)&sA[buf ^ 1][0]);
                __builtin_amdgcn_s_wait_tensorcnt(1);
            } else {
                __builtin_amdgcn_s_wait_tensorcnt(0);
            }
        }
        const float* sAc = &sA[buf][0];
#else
        // Manual staging: BM x KC floats = 512 float4, 2 per thread.
#pragma unroll
        for (int it = 0; it < (BM * KC) / (4 * 256); ++it) {
            int u = tid + it * 256;
            int r = u >> 3;
            int c4 = (u & 7) << 2;
            float4 v = *(const float4*)&A[(size_t)(rowBase + r) * kN + k0 + c4];
            *(float4*)&sA[0][r * LDA_S + c4] = v;
        }
        const float* sAc = &sA[0][0];
        if (k0 + KC < kN)
            __builtin_prefetch(&A[(size_t)(rowBase + (tid >> 2)) * kN + k0 + KC], 0, 0);
#endif
        __syncthreads();

#pragma unroll
        for (int kk = 0; kk < KC; kk += 4) {
            // A fragment 16x4: lanes 0-15 = rows, VGPR pair = K{0,1} / K{2,3}.
            v2f afrag = *(const v2f*)&sAc[(rw + lm) * LDA_S + kk + lhi * 2];
#pragma unroll
            for (int ct = 0; ct < 2; ++ct) {
                // B fragment 4x16, transposed layout -> one contiguous b64.
                v2f bfrag = *(const v2f*)&sBt[(ch + ct * 16 + lm) * LDKT + kk + lhi * 2];
                acc[ct] = __builtin_amdgcn_wmma_f32_16x16x4_f32(
                    false, afrag, false, bfrag, (short)0, acc[ct], false, false);
            }
        }
        __syncthreads();
    }

    // Fused epilogue: D-normalize rows + residual.
    const float resid = residp[0];
#pragma unroll
    for (int ct = 0; ct < 2; ++ct) {
#pragma unroll
        for (int v = 0; v < 8; ++v) {
            int grow = rowBase + rw + v + lhi * 8;   // C/D: VGPR v -> M = v (+8)
            int gcol = nBase + ch + ct * 16 + lm;    // lane(0-15) -> N
            C[(size_t)grow * NCOLS + gcol] =
                drow[grow] * acc[ct][v] + resid * X0[(size_t)grow * NCOLS + gcol];
        }
    }
}

// ------------------- K6: hid = out1 @ W1 + b1 ; hs = dcol (.) hid ----------
__global__ __launch_bounds__(256)
void hid_kernel(const float* __restrict__ out1, const float* __restrict__ W1,
                const float* __restrict__ b1, const float* __restrict__ dcol,
                float* __restrict__ hid, float* __restrict__ hs) {
    __shared__ float srow[2 * kIN];
    const int tid = threadIdx.x;
    const int rowBase = blockIdx.x * 2;
    srow[tid]       = out1[(size_t)rowBase * kIN + tid];
    srow[tid + 256] = out1[(size_t)rowBase * kIN + 256 + tid];
    __syncthreads();
    const int r = tid >> 7;
    const int h = tid & 127;
    const float* sr = &srow[r * kIN];
    float sum = b1[h];
    for (int k = 0; k < kIN; ++k) sum = fmaf(sr[k], W1[k * kHID + h], sum);
    const int row = rowBase + r;
    hid[(size_t)row * kHID + h] = sum;
    hs[(size_t)row * kHID + h] = dcol[row] * sum;
}

// --------------------------- K8: classifier --------------------------------
__global__ __launch_bounds__(256)
void out_kernel(const float* __restrict__ out2, const float* __restrict__ W2,
                const float* __restrict__ b2, float* __restrict__ out) {
    int i = blockIdx.x * 256 + threadIdx.x;
    if (i >= kN) return;
    float s0 = b2[0], s1 = b2[1];
    for (int k = 0; k < kHID; ++k) {
        float v = out2[(size_t)i * kHID + k];
        s0 = fmaf(v, W2[2 * k + 0], s0);
        s1 = fmaf(v, W2[2 * k + 1], s1);
    }
    out[2 * i + 0] = s0;
    out[2 * i + 1] = s1;
}

// ---------------------------------------------------------------------------
extern "C" void kernel_launch(void* const* d_in, const int* in_sizes, int n_in,
                              void* d_out, int out_size, void* d_ws, size_t ws_size,
                              hipStream_t stream) {
    const float* x    = (const float*)d_in[0];
    const float* adj  = (const float*)d_in[1];
    const float* xdeg = (const float*)d_in[2];
    const float* ydeg = (const float*)d_in[3];
    const float* We1  = (const float*)d_in[4];
    const float* be1  = (const float*)d_in[5];
    const float* We2  = (const float*)d_in[6];
    const float* be2  = (const float*)d_in[7];
    const float* W1   = (const float*)d_in[8];
    const float* b1   = (const float*)d_in[9];
    const float* W2   = (const float*)d_in[10];
    const float* b2   = (const float*)d_in[11];
    const float* resid = (const float*)d_in[12];
    float* out = (float*)d_out;

    // Workspace carve-up (all fully rewritten each call).
    float* ws   = (float*)d_ws;
    float* A    = ws;                            // kN*kN
    float* rs   = A + (size_t)kN * kN;           // kN
    float* cs   = rs + kN;                       // kN
    float* drow = cs + kN;                       // kN
    float* dcol = drow + kN;                     // kN
    float* degr = dcol + kN;                     // kN
    float* degc = degr + kN;                     // kN
    float* xs   = degc + kN;                     // kN*kIN
    float* out1 = xs + (size_t)kN * kIN;         // kN*kIN
    float* hid  = out1 + (size_t)kN * kIN;       // kN*kHID
    float* hs   = hid + (size_t)kN * kHID;       // kN*kHID
    float* out2 = hs + (size_t)kN * kHID;        // kN*kHID

    deg_kernel<<<kN / 256, 256, 0, stream>>>(xdeg, ydeg, degr, degc, cs);
    build_A_kernel<<<kN, 256, 0, stream>>>(adj, degr, degc, We1, be1, We2, be2, A, rs);
    colsum_kernel<<<dim3(kN / 256, 32), 256, 0, stream>>>(A, cs);
    norm_kernel<<<kN / 256, 256, 0, stream>>>(rs, cs, drow, dcol);
    scalex_kernel<<<(kN * kIN) / 256, 256, 0, stream>>>(x, dcol, xs);
    gemm_prop_kernel<kIN><<<dim3(kIN / BN, kN / BM), 256, 0, stream>>>(
        A, xs, x, drow, resid, out1);
    hid_kernel<<<kN / 2, 256, 0, stream>>>(out1, W1, b1, dcol, hid, hs);
    gemm_prop_kernel<kHID><<<dim3(kHID / BN, kN / BM), 256, 0, stream>>>(
        A, hs, hid, drow, resid, out2);
    out_kernel<<<kN / 256, 256, 0, stream>>>(out2, W2, b2, out);
}